// LSSTransform_1022202217008
// MI455X (gfx1250) — compile-verified
//
#include <hip/hip_runtime.h>
#include <hip/hip_bf16.h>

typedef float v2f __attribute__((ext_vector_type(2)));
typedef float v8f __attribute__((ext_vector_type(8)));

#define NIMG   24      // B*N
#define CIN    512
#define DBINS  41
#define CT     64
#define MROWS  105     // DBINS + CT
#define PIX    704     // FH*FW = 16*44
#define FW_    44
#define NTILES 44      // PIX/16

// ---------------- Kernel 0: per-image geometry precompute -------------------
__device__ inline void inv3x3(const float* a, float* o) {
    float c00 =  a[4]*a[8] - a[5]*a[7];
    float c01 = -(a[3]*a[8] - a[5]*a[6]);
    float c02 =  a[3]*a[7] - a[4]*a[6];
    float det = a[0]*c00 + a[1]*c01 + a[2]*c02;
    float id  = 1.0f / det;
    o[0] = c00 * id;
    o[1] = -(a[1]*a[8] - a[2]*a[7]) * id;
    o[2] =  (a[1]*a[5] - a[2]*a[4]) * id;
    o[3] = c01 * id;
    o[4] =  (a[0]*a[8] - a[2]*a[6]) * id;
    o[5] = -(a[0]*a[5] - a[2]*a[3]) * id;
    o[6] = c02 * id;
    o[7] = -(a[0]*a[7] - a[1]*a[6]) * id;
    o[8] =  (a[0]*a[4] - a[1]*a[3]) * id;
}

__global__ void lss_geo_kernel(const float* __restrict__ rots,
                               const float* __restrict__ trans,
                               const float* __restrict__ intrins,
                               const float* __restrict__ post_rots,
                               const float* __restrict__ post_trans,
                               float* __restrict__ geo) {
    int img = blockIdx.x * blockDim.x + threadIdx.x;
    if (img >= NIMG) return;
    float R[9], K[9], P[9], invK[9], invP[9], comb[9];
    for (int i = 0; i < 9; ++i) {
        R[i] = rots[img*9 + i];
        K[i] = intrins[img*9 + i];
        P[i] = post_rots[img*9 + i];
    }
    inv3x3(K, invK);
    inv3x3(P, invP);
    for (int r = 0; r < 3; ++r)
        for (int c = 0; c < 3; ++c)
            comb[r*3 + c] = R[r*3+0]*invK[0*3+c] + R[r*3+1]*invK[1*3+c] + R[r*3+2]*invK[2*3+c];
    float* g = geo + img*24;
    for (int i = 0; i < 9; ++i) { g[i] = invP[i]; g[9 + i] = comb[i]; }
    for (int i = 0; i < 3; ++i) { g[18 + i] = trans[img*3 + i]; g[21 + i] = post_trans[img*3 + i]; }
}

// ------- Kernel 1: WMMA GEMM (Wd @ X) + bias + softmax(depth) -------------
// grid (NTILES, NIMG), block 224 threads = 7 waves; wave w owns M-tile w.
__global__ __launch_bounds__(224) void lss_gemm_softmax_kernel(
        const float* __restrict__ x, const float* __restrict__ Wd,
        const float* __restrict__ bd,
        float* __restrict__ depth_out,   // [img][41][704]
        float* __restrict__ cvt_out)     // [img][704][64] channel-last
{
    const int tileN = blockIdx.x;
    const int img   = blockIdx.y;
    const int tid   = threadIdx.x;
    const int wave  = tid >> 5;          // 0..6 -> M tile
    const int lane  = tid & 31;
    const int pixbase = tileN * 16;

    __shared__ __align__(16) float Xs[CIN * 16];   // [k][col], 32 KB
    __shared__ float feat[112 * 17];               // [row][col], padded
    __shared__ float mx[16], ism[16];

    // Cooperative load: 512x16 strip of X as float4 (b128 loads)
    {
        const float4* src = reinterpret_cast<const float4*>(
            x + (size_t)img * CIN * PIX + pixbase);
        for (int i = tid; i < CIN * 4; i += 224) {
            int k = i >> 2, cv = i & 3;
            float4 v = src[(size_t)k * (PIX / 4) + cv];
            *reinterpret_cast<float4*>(&Xs[k * 16 + cv * 4]) = v;
        }
    }
    __syncthreads();

    // A fragment (16x4 f32 layout): lane holds row M = lane&15, K = koff..koff+1
    // Rows >= MROWS compute garbage that is never stored (row m of C depends
    // only on row m of A), so no masking needed -- just keep loads in-bounds.
    const int col  = lane & 15;
    const int m    = wave * 16 + col;
    const int koff = (lane >> 4) << 1;                 // 0 or 2
    const int mClamped = (m < MROWS) ? m : (MROWS - 1);
    const float2* aPtr = reinterpret_cast<const float2*>(
        Wd + (size_t)mClamped * CIN + koff);

    v8f acc = {};
#pragma unroll 4
    for (int k0 = 0; k0 < CIN; k0 += 4) {
        float2 av = aPtr[k0 >> 1];
        v2f a; a.x = av.x; a.y = av.y;
        int kb = k0 + koff;
        v2f b; b.x = Xs[kb * 16 + col]; b.y = Xs[(kb + 1) * 16 + col];
        acc = __builtin_amdgcn_wmma_f32_16x16x4_f32(
            /*neg_a=*/false, a, /*neg_b=*/false, b,
            /*c_mod=*/(short)0, acc, /*reuse_a=*/false, /*reuse_b=*/false);
    }

    // C fragment -> LDS with bias (row = r + (lane<16 ? 0 : 8)).
    // Store unconditionally (feat has 112 rows; rows >= 105 never read);
    // only clamp the bias index to stay in-bounds on bd.
    {
        int rbase = wave * 16 + ((lane < 16) ? 0 : 8);
#pragma unroll
        for (int r = 0; r < 8; ++r) {
            int row  = rbase + r;
            int brow = (row < MROWS) ? row : (MROWS - 1);
            feat[row * 17 + col] = acc[r] + bd[brow];
        }
    }
    __syncthreads();

    // Per-column softmax stats over depth rows 0..40
    if (tid < 16) {
        float m0 = -3.402823466e+38f;
        for (int d = 0; d < DBINS; ++d) m0 = fmaxf(m0, feat[d * 17 + tid]);
        float s = 0.0f;
        for (int d = 0; d < DBINS; ++d) s += __expf(feat[d * 17 + tid] - m0);
        mx[tid] = m0; ism[tid] = 1.0f / s;
    }
    __syncthreads();

    // depth out: [img][d][pix]
    for (int i = tid; i < DBINS * 16; i += 224) {
        int d = i >> 4, c2 = i & 15;
        depth_out[((size_t)img * DBINS + d) * PIX + pixbase + c2] =
            __expf(feat[d * 17 + c2] - mx[c2]) * ism[c2];
    }
    // cvt out: [img][pix][c] (channel-last for coalesced scatter loads)
    for (int i = tid; i < 16 * CT; i += 224) {
        int c2 = i >> 6, c = i & 63;
        cvt_out[((size_t)img * PIX + pixbase + c2) * CT + c] = feat[(DBINS + c) * 17 + c2];
    }
}

// ---------- Kernel 2: voxelize + scatter-add (wave per point) --------------
__global__ __launch_bounds__(256) void lss_scatter_kernel(
        const float* __restrict__ depth, const float* __restrict__ cvt,
        const float* __restrict__ geo, float* __restrict__ out)
{
    const int NPTS = NIMG * DBINS * PIX;
    int gid = blockIdx.x * 256 + threadIdx.x;
    int point = gid >> 5;
    if (point >= NPTS) return;
    int lane = gid & 31;

    int pix = point % PIX;
    int t   = point / PIX;
    int d   = t % DBINS;
    int img = t / DBINS;
    int b   = img / 6;
    int h = pix / FW_, w = pix % FW_;

    const float* g = geo + img * 24;
    // frustum point
    float fx = (float)w * (703.0f / 43.0f);
    float fy = (float)h * 17.0f;
    float fd = 4.0f + (float)d;
    // undo post transform
    float px = fx - g[21], py = fy - g[22], pz = fd - g[23];
    float qx = g[0]*px + g[1]*py + g[2]*pz;
    float qy = g[3]*px + g[4]*py + g[5]*pz;
    float qz = g[6]*px + g[7]*py + g[8]*pz;
    qx *= qz; qy *= qz;
    // cam -> ego
    float rx = g[9]*qx  + g[10]*qy + g[11]*qz + g[18];
    float ry = g[12]*qx + g[13]*qy + g[14]*qz + g[19];
    float rz = g[15]*qx + g[16]*qy + g[17]*qz + g[20];
    // voxelize (trunc toward zero, matching astype(int32))
    int gx = (int)((rx + 51.2f) / 0.8f);
    int gy = (int)((ry + 51.2f) / 0.8f);
    int gz = (int)((rz + 10.0f) / 20.0f);
    if (gx < 0 || gx >= 128 || gy < 0 || gy >= 128 || gz != 0) return;

    float dep = depth[((size_t)img * DBINS + d) * PIX + pix];
    const float* cp = cvt + ((size_t)img * PIX + pix) * CT;
    float v0 = dep * cp[lane];
    float v1 = dep * cp[lane + 32];

    size_t ob = (size_t)(b * CT) * 16384 + (size_t)(gy * 128 + gx);
    unsafeAtomicAdd(out + ob + (size_t)lane * 16384, v0);
    unsafeAtomicAdd(out + ob + (size_t)(lane + 32) * 16384, v1);
}

// ---------------------------------------------------------------------------
extern "C" void kernel_launch(void* const* d_in, const int* in_sizes, int n_in,
                              void* d_out, int out_size, void* d_ws, size_t ws_size,
                              hipStream_t stream) {
    const float* x          = (const float*)d_in[0];
    const float* rots       = (const float*)d_in[1];
    const float* trans      = (const float*)d_in[2];
    const float* intrins    = (const float*)d_in[3];
    const float* post_rots  = (const float*)d_in[4];
    const float* post_trans = (const float*)d_in[5];
    const float* Wd         = (const float*)d_in[6];
    const float* bd         = (const float*)d_in[7];
    float* out = (float*)d_out;

    char* ws = (char*)d_ws;
    float* geo   = (float*)(ws);                       // 24*24*4 = 2304 B
    float* depth = (float*)(ws + 4096);                // 24*41*704*4 = 2,770,944 B
    float* cvt   = (float*)(ws + 4096 + 2770944);      // 24*704*64*4 = 4,325,376 B

    hipMemsetAsync(d_out, 0, (size_t)out_size * sizeof(float), stream);

    lss_geo_kernel<<<1, 32, 0, stream>>>(rots, trans, intrins, post_rots, post_trans, geo);

    lss_gemm_softmax_kernel<<<dim3(NTILES, NIMG), 224, 0, stream>>>(
        x, Wd, bd, depth, cvt);

    int nThreads = NIMG * DBINS * PIX * 32;
    lss_scatter_kernel<<<(nThreads + 255) / 256, 256, 0, stream>>>(depth, cvt, geo, out);
}